// Spike_Convolve_16097537426001
// MI455X (gfx1250) — compile-verified
//
#include <hip/hip_runtime.h>
#include <hip/hip_bf16.h>

// ---------------------------------------------------------------------------
// Spike projection + causal alpha-kernel convolution for MI455X (gfx1250).
//   Pass 1: build alpha-function kernel bank  kern[s][c][t]   (tiny)
//   Pass 2: in_e^T = (S_e @ C_syn_e^T)^T via V_WMMA_F32_16X16X4_F32
//           in_i^T = (S_i @ C_syn_i^T)^T      (stored time-major per subunit)
//   Pass 3: out[t,s] = sum_k ke[s,k]*in_e[t-k,s] + ki[s,k]*in_i[t-k,s]
// ---------------------------------------------------------------------------

typedef __attribute__((ext_vector_type(2))) float v2f;
typedef __attribute__((ext_vector_type(8))) float v8f;

#define TSYN 201
#define BSTRIDE 502     // LDS row stride (floats) for B staging; 502 % 64 != 0 -> no bank wall

// ---------------- Pass 1: kernel bank --------------------------------------
__global__ __launch_bounds__(256) void make_kern(const float* __restrict__ K_syn,
                                                 const float* __restrict__ tau_syn,
                                                 const float* __restrict__ delta_syn,
                                                 float* __restrict__ kern,
                                                 int sub, int B) {
    int i = blockIdx.x * 256 + threadIdx.x;       // over sub*2*TSYN
    int total = sub * 2 * TSYN;
    if (i >= total) return;
    int t = i % TSYN;
    int c = (i / TSYN) & 1;
    int s = i / (TSYN * 2);
    float ts = (float)t - delta_syn[s * 2 + c];
    ts = ts < 0.0f ? 0.0f : ts;
    float acc = 0.0f;
    for (int b = 0; b < B; ++b) {
        float tau = expf(tau_syn[b * 2 + c]);
        float tt  = ts / tau;
        acc = fmaf(K_syn[(s * B + b) * 2 + c], tt * expf(-tt), acc);
    }
    kern[i] = acc;                                 // layout [s][c][t]
}

// ---------------- Pass 2: projection GEMM with f32 WMMA --------------------
// A: [T x K] row-major (spikes).  Bmat: [sub x K] row-major (C_syn, == B^T).
// Ct: [sub x T] (transposed output) so the conv pass reads time-major.
// Requires K % 4 == 0, K <= 500, sub <= 32.  Wave handles a 16(M) x 32(N) tile.
__global__ __launch_bounds__(256) void gemm_proj(const float* __restrict__ A,
                                                 const float* __restrict__ Bmat,
                                                 float* __restrict__ Ct,
                                                 int T, int K, int sub) {
    __shared__ float blds[32 * BSTRIDE];           // 64,256 B

    const int tid = threadIdx.x;
    // Stage B into LDS as blds[n][k] (zero-pad n in [sub,32)).
    for (int idx = tid; idx < 32 * K; idx += 256) {
        int n = idx / K;
        int k = idx - n * K;
        blds[n * BSTRIDE + k] = (n < sub) ? Bmat[n * K + k] : 0.0f;
    }
    __syncthreads();

    const int wave  = tid >> 5;
    const int lane  = tid & 31;
    const int row   = lane & 15;                   // M row within tile / N col of B frag
    const int khalf = lane >> 4;                   // 0 -> K {0,1}, 1 -> K {2,3}
    long m0 = ((long)blockIdx.x * 8 + wave) * 16;
    if (m0 >= T) return;                           // wave-uniform exit (EXEC stays all-1s)

    long mrow = m0 + row;
    if (mrow >= T) mrow = T - 1;                   // clamp: keeps EXEC full; rows >=T never stored

    const float* aptr = A + mrow * (long)K + 2 * khalf;
    const float* bl0  = &blds[row * BSTRIDE + 2 * khalf];          // N-tile 0 (cols 0..15)
    const float* bl1  = &blds[(16 + row) * BSTRIDE + 2 * khalf];   // N-tile 1 (cols 16..31)

    v8f c0 = {};
    v8f c1 = {};

    int kk = 0;
    for (; kk + 32 <= K; kk += 32) {
        __builtin_prefetch(aptr + kk + 256, 0, 3);                 // global_prefetch_b8
#pragma unroll
        for (int u = 0; u < 32; u += 4) {
            v2f a  = *(const v2f*)(aptr + kk + u);
            v2f b0 = *(const v2f*)(bl0 + kk + u);
            v2f b1 = *(const v2f*)(bl1 + kk + u);
            c0 = __builtin_amdgcn_wmma_f32_16x16x4_f32(false, a, false, b0, (short)0, c0, false, false);
            c1 = __builtin_amdgcn_wmma_f32_16x16x4_f32(false, a, false, b1, (short)0, c1, false, false);
        }
    }
    for (; kk < K; kk += 4) {
        v2f a  = *(const v2f*)(aptr + kk);
        v2f b0 = *(const v2f*)(bl0 + kk);
        v2f b1 = *(const v2f*)(bl1 + kk);
        c0 = __builtin_amdgcn_wmma_f32_16x16x4_f32(false, a, false, b0, (short)0, c0, false, false);
        c1 = __builtin_amdgcn_wmma_f32_16x16x4_f32(false, a, false, b1, (short)0, c1, false, false);
    }

    // C/D layout: VGPR r -> lanes 0-15: (M=r, N=lane); lanes 16-31: (M=r+8, N=lane-16)
#pragma unroll
    for (int r = 0; r < 8; ++r) {
        long m = m0 + r + 8 * khalf;
        if (m < T) {
            if (row < sub)       Ct[(long)row * T + m]        = c0[r];
            if (16 + row < sub)  Ct[(long)(16 + row) * T + m] = c1[r];
        }
    }
}

// ---------------- Pass 3: depthwise causal conv ----------------------------
// Grid: x over time tiles of 256, y = subunit. LDS-staged signal window + taps.
__global__ __launch_bounds__(256) void conv_kernel(const float* __restrict__ ine,
                                                   const float* __restrict__ ini,
                                                   const float* __restrict__ kern,
                                                   float* __restrict__ out,
                                                   int T, int sub) {
    __shared__ float ke[TSYN];
    __shared__ float ki[TSYN];
    __shared__ float se[256 + TSYN - 1];
    __shared__ float si[256 + TSYN - 1];

    const int s  = blockIdx.y;
    const long t0 = (long)blockIdx.x * 256;
    const int tid = threadIdx.x;

    for (int i = tid; i < TSYN; i += 256) {
        ke[i] = kern[(s * 2 + 0) * TSYN + i];
        ki[i] = kern[(s * 2 + 1) * TSYN + i];
    }
    for (int i = tid; i < 256 + TSYN - 1; i += 256) {
        long g = t0 - (TSYN - 1) + i;
        bool ok = (g >= 0) && (g < T);
        se[i] = ok ? ine[(long)s * T + g] : 0.0f;
        si[i] = ok ? ini[(long)s * T + g] : 0.0f;
    }
    __syncthreads();

    float acc = 0.0f;
#pragma unroll 4
    for (int k = 0; k < TSYN; ++k) {
        int idx = tid + (TSYN - 1) - k;            // time t - k within window
        acc = fmaf(ke[k], se[idx], acc);
        acc = fmaf(ki[k], si[idx], acc);
    }
    long t = t0 + tid;
    if (t < T) out[t * (long)sub + s] = acc;
}

// ---------------------------------------------------------------------------
extern "C" void kernel_launch(void* const* d_in, const int* in_sizes, int n_in,
                              void* d_out, int out_size, void* d_ws, size_t ws_size,
                              hipStream_t stream) {
    const float* S_e       = (const float*)d_in[0];
    const float* S_i       = (const float*)d_in[1];
    const float* C_syn_e   = (const float*)d_in[2];
    const float* C_syn_i   = (const float*)d_in[3];
    const float* K_syn     = (const float*)d_in[4];
    const float* tau_syn   = (const float*)d_in[5];
    const float* delta_syn = (const float*)d_in[6];
    float* out = (float*)d_out;

    const int sub = in_sizes[6] / 2;               // delta_syn [sub, 2]
    const int n_e = in_sizes[2] / sub;             // C_syn_e [sub, n_e]
    const int n_i = in_sizes[3] / sub;             // C_syn_i [sub, n_i]
    const int T   = in_sizes[0] / n_e;             // S_e [T, n_e]
    const int B   = in_sizes[4] / (sub * 2);       // K_syn [sub, B, 2]

    // Workspace layout: [kern | in_e^T | in_i^T]
    char*  ws   = (char*)d_ws;
    float* kern = (float*)ws;                                  // sub*2*TSYN floats
    float* ine  = (float*)(ws + 65536);                        // sub*T floats
    float* ini  = ine + (size_t)sub * (size_t)T;               // sub*T floats

    const int ktot = sub * 2 * TSYN;
    make_kern<<<(ktot + 255) / 256, 256, 0, stream>>>(K_syn, tau_syn, delta_syn, kern, sub, B);

    const int gblocks = (T + 127) / 128;           // 8 waves/block * 16 rows/wave
    gemm_proj<<<gblocks, 256, 0, stream>>>(S_e, C_syn_e, ine, T, n_e, sub);
    gemm_proj<<<gblocks, 256, 0, stream>>>(S_i, C_syn_i, ini, T, n_i, sub);

    conv_kernel<<<dim3((T + 255) / 256, sub), 256, 0, stream>>>(ine, ini, kern, out, T, sub);
}